// FistaQuantize_62586263437525
// MI455X (gfx1250) — compile-verified
//
#include <hip/hip_runtime.h>
#include <math.h>

// ---------------- problem constants ----------------
#define DIM     128
#define KATOMS  1024
#define NPATCH  16384            // 16 * 32 * 32
#define ALPHA   0.1f
#define NSTEPS  50
#define TOPK    10
#define KN      ((size_t)KATOMS * (size_t)NPATCH)   // 16,777,216

// output layout (floats, concatenated in reference return order)
#define OFF_OUT0  0
#define OFF_QDE   2097152
#define OFF_QDD   2097153
#define OFF_IDS   2097154
#define OFF_STEPS 18874370
#define OFF_MEAND 18874371
#define OFF_MEANZ 18874372
#define OFF_NORMZ 18874373
#define OFF_TOPP  18874374
#define OFF_NZERO 18874375

// device scalar slots in workspace
#define S_L     0
#define S_STEP  1
#define S_THR   2
#define S_QD    3
#define S_ABSD  4
#define S_ABSZ  5
#define S_SUML0 6

// GEMM tiling: 128x64 block tile, K-slab 32, 256 threads (8 waves),
// each wave computes a 16x64 strip = 4 x (16x16) WMMA accumulators.
// LDS tiles are k-PAIR-INTERLEAVED: element (k, m) lives at
//   [pair = k>>1][2*m + (k&1)]  -> a WMMA fragment (k0,k0+1) is ONE b64 load.
#define BM 128
#define BN 64
#define BK 32
#define NPAIR (BK / 2)           // 16 pair-rows per slab
#define SA (2 * BM + 2)          // 258 floats: 256-dword row + 2-dword pad (even => b64 aligned)
#define SB (2 * BN + 2)          // 130 floats: 128-dword row + 2-dword pad

typedef __attribute__((ext_vector_type(2))) float v2f;
typedef __attribute__((ext_vector_type(8))) float v8f;
typedef unsigned int v4u __attribute__((ext_vector_type(4)));
typedef int v8i __attribute__((ext_vector_type(8)));
typedef int v4i __attribute__((ext_vector_type(4)));

// ---------------- WMMA inner product on interleaved LDS tiles ----------------
// FP32 WMMA 16x16x4 (ISA 7.12.2): lane l: m = l&15, k-pair = (l>>4); the two
// k values of a fragment are adjacent floats in LDS -> single ds_load_b64.
__device__ __forceinline__ void wmma_tile2(const float* __restrict__ As2,
                                           const float* __restrict__ Bs2,
                                           v8f* acc, int wrow, int lane) {
  const int m  = wrow + (lane & 15);
  const int nb = lane & 15;
  const int kp = lane >> 4;
#pragma unroll
  for (int k = 0; k < BK; k += 4) {
    const int pr = (k >> 1) + kp;
    v2f a = *reinterpret_cast<const v2f*>(&As2[pr * SA + 2 * m]);
#pragma unroll
    for (int s = 0; s < 4; ++s) {
      v2f b = *reinterpret_cast<const v2f*>(&Bs2[pr * SB + 2 * (s * 16 + nb)]);
      acc[s] = __builtin_amdgcn_wmma_f32_16x16x4_f32(
          false, a, false, b, (short)0, acc[s], false, false);
    }
  }
}

// interleave two k-rows (4 columns each) into pair layout: 4x ds_store_b64
__device__ __forceinline__ void store_pair4(float* dst, float4 a, float4 b) {
  reinterpret_cast<float2*>(dst)[0] = make_float2(a.x, b.x);
  reinterpret_cast<float2*>(dst)[1] = make_float2(a.y, b.y);
  reinterpret_cast<float2*>(dst)[2] = make_float2(a.z, b.z);
  reinterpret_cast<float2*>(dst)[3] = make_float2(a.w, b.w);
}

// ---------------- Tensor Data Mover descriptor helpers (ISA ch.8) ----------------
// group0: [1:0]=count=1, [63:32]=lds byte addr, [120:64]=global byte addr, [127:126]=type=2
__device__ __forceinline__ v4u tdm_group0(unsigned lds_addr, unsigned long long gaddr) {
  v4u g;
  g.x = 1u;
  g.y = lds_addr;
  g.z = (unsigned)(gaddr & 0xFFFFFFFFull);
  g.w = (unsigned)((gaddr >> 32) & 0x1FFFFFFull) | 0x80000000u;
  return g;
}
// group1: data_size=4B (code 2); pad_enable with per-row LDS padding; 2D tile.
// dims/strides in elements; tile_dim1 = rows, tensor_dim0_stride = row pitch.
__device__ __forceinline__ v8i tdm_group1(unsigned td0, unsigned td1,
                                          unsigned tile0, unsigned tile1,
                                          unsigned long long stride0,
                                          unsigned pint, unsigned pamt) {
  v8i g;
  g[0] = (int)((2u << 16) | (1u << 20) | (pint << 22) | (pamt << 25));
  g[1] = (int)((td0 & 0xFFFFu) << 16);
  g[2] = (int)(((td0 >> 16) & 0xFFFFu) | ((td1 & 0xFFFFu) << 16));
  g[3] = (int)(((td1 >> 16) & 0xFFFFu) | ((tile0 & 0xFFFFu) << 16));
  g[4] = (int)(tile1 & 0xFFFFu);
  g[5] = (int)(stride0 & 0xFFFFFFFFull);
  g[6] = (int)((stride0 >> 32) & 0xFFFFull);
  g[7] = 0;
  return g;
}

// ---------------- utility kernels ----------------
__global__ void zero_f32(float* __restrict__ p, size_t n) {
  size_t i = (size_t)blockIdx.x * blockDim.x + threadIdx.x;
  size_t stride = (size_t)gridDim.x * blockDim.x;
  for (; i < n; i += stride) p[i] = 0.0f;
}

// (B,C,H,W) -> patch-major (N, DIM)
__global__ __launch_bounds__(256) void permute_in(const float* __restrict__ x,
                                                  float* __restrict__ Xf) {
  size_t t = (size_t)blockIdx.x * 256 + threadIdx.x;
  if (t >= (size_t)NPATCH * DIM) return;
  int d = (int)(t & (DIM - 1));
  int n = (int)(t >> 7);
  int b = n >> 10, p = n & 1023;
  Xf[t] = x[(size_t)b * (DIM * 1024) + (size_t)d * 1024 + p];
}

// unit-norm columns of D (d x K); writes Dn (d-major) and DnT (K-major),
// accumulates sum |Dn| for mean_D.
__global__ __launch_bounds__(256) void normalize_dict(const float* __restrict__ D,
                                                      float* __restrict__ Dn,
                                                      float* __restrict__ DnT,
                                                      float* __restrict__ scal) {
  int k = blockIdx.x * 256 + threadIdx.x;
  if (k >= KATOMS) return;
  float s = 0.0f;
  for (int d = 0; d < DIM; ++d) { float v = D[(size_t)d * KATOMS + k]; s += v * v; }
  float inv = rsqrtf(s);
  float sa = 0.0f;
  for (int d = 0; d < DIM; ++d) {
    float v = D[(size_t)d * KATOMS + k] * inv;
    Dn[(size_t)d * KATOMS + k] = v;
    DnT[(size_t)k * DIM + d]   = v;
    sa += fabsf(v);
  }
  atomicAdd(&scal[S_ABSD], sa);
}

// ---------------- G = Dn^T Dn (M=N=1024, inner=128) ----------------
// Writes row-major G (for power iteration) AND pair-interleaved G2
// (element (m,k) -> G2[(k>>1)*2*KATOMS + 2*m + (k&1)]) consumed by the TDM.
__global__ __launch_bounds__(256) void gemm_gram(const float* __restrict__ Dn,
                                                 float* __restrict__ G,
                                                 float* __restrict__ G2) {
  __shared__ float As2[NPAIR * SA];
  __shared__ float Bs2[NPAIR * SB];
  const int tid   = threadIdx.x;
  const int tileM = (blockIdx.x & 7) * BM;
  const int tileN = (blockIdx.x >> 3) * BN;
  const int lane  = tid & 31, wave = tid >> 5;
  v8f acc[4] = {};
  for (int kk = 0; kk < DIM; kk += BK) {
#pragma unroll
    for (int i = 0; i < 2; ++i) {                      // A: Dn rows 2dp,2dp+1 -> pair rows
      int it = tid + i * 256;
      int dp = it >> 5, mg = (it & 31) << 2;
      float4 r0 = *reinterpret_cast<const float4*>(
          &Dn[(size_t)(kk + 2 * dp) * KATOMS + tileM + mg]);
      float4 r1 = *reinterpret_cast<const float4*>(
          &Dn[(size_t)(kk + 2 * dp + 1) * KATOMS + tileM + mg]);
      store_pair4(&As2[dp * SA + 2 * mg], r0, r1);
    }
    {                                                  // B: same, 64-wide
      int dp = tid >> 4, ng = (tid & 15) << 2;
      float4 r0 = *reinterpret_cast<const float4*>(
          &Dn[(size_t)(kk + 2 * dp) * KATOMS + tileN + ng]);
      float4 r1 = *reinterpret_cast<const float4*>(
          &Dn[(size_t)(kk + 2 * dp + 1) * KATOMS + tileN + ng]);
      store_pair4(&Bs2[dp * SB + 2 * ng], r0, r1);
    }
    __syncthreads();
    wmma_tile2(As2, Bs2, acc, wave * 16, lane);
    __syncthreads();
  }
  const int mh = (lane >> 4) << 3;
#pragma unroll
  for (int s = 0; s < 4; ++s)
#pragma unroll
    for (int r = 0; r < 8; ++r) {
      int gm = tileM + wave * 16 + r + mh;
      int gn = tileN + s * 16 + (lane & 15);
      float c = acc[s][r];
      G[(size_t)gm * KATOMS + gn] = c;
      G2[(size_t)(gn >> 1) * (2 * KATOMS) + 2 * gm + (gn & 1)] = c;
    }
}

// ---------------- DtX = Dn^T Xf^T (M=1024, N=16384, inner=128) ----------------
__global__ __launch_bounds__(256) void gemm_dtx(const float* __restrict__ Dn,
                                                const float* __restrict__ Xf,
                                                float* __restrict__ DtX) {
  __shared__ float As2[NPAIR * SA];
  __shared__ float Bs2[NPAIR * SB];
  const int tid   = threadIdx.x;
  const int tileM = (blockIdx.x & 7) * BM;
  const int tileN = (blockIdx.x >> 3) * BN;
  const int lane  = tid & 31, wave = tid >> 5;
  v8f acc[4] = {};
  for (int kk = 0; kk < DIM; kk += BK) {
#pragma unroll
    for (int i = 0; i < 2; ++i) {                      // A: Dn pair-interleave
      int it = tid + i * 256;
      int dp = it >> 5, mg = (it & 31) << 2;
      float4 r0 = *reinterpret_cast<const float4*>(
          &Dn[(size_t)(kk + 2 * dp) * KATOMS + tileM + mg]);
      float4 r1 = *reinterpret_cast<const float4*>(
          &Dn[(size_t)(kk + 2 * dp + 1) * KATOMS + tileM + mg]);
      store_pair4(&As2[dp * SA + 2 * mg], r0, r1);
    }
#pragma unroll
    for (int i = 0; i < 2; ++i) {                      // B: Xf[n][d] d-contig -> pair layout
      int it = tid + i * 256;
      int nr = it >> 3, dc = (it & 7) << 2;
      float4 v = *reinterpret_cast<const float4*>(
          &Xf[(size_t)(tileN + nr) * DIM + kk + dc]);
      *reinterpret_cast<float2*>(&Bs2[(dc >> 1) * SB + 2 * nr])       = make_float2(v.x, v.y);
      *reinterpret_cast<float2*>(&Bs2[((dc >> 1) + 1) * SB + 2 * nr]) = make_float2(v.z, v.w);
    }
    __syncthreads();
    wmma_tile2(As2, Bs2, acc, wave * 16, lane);
    __syncthreads();
  }
  const int mh = (lane >> 4) << 3;
#pragma unroll
  for (int s = 0; s < 4; ++s)
#pragma unroll
    for (int r = 0; r < 8; ++r) {
      int gm = tileM + wave * 16 + r + mh;
      int gn = tileN + s * 16 + (lane & 15);
      DtX[(size_t)gm * NPATCH + gn] = acc[s][r];
    }
}

// ---------------- power iteration (single block, G symmetric) ----------------
__device__ __forceinline__ float block_reduce_sum_1024(float v, float* red) {
#pragma unroll
  for (int off = 16; off > 0; off >>= 1) v += __shfl_xor(v, off, 32);
  const int lane = threadIdx.x & 31, w = threadIdx.x >> 5;
  if (lane == 0) red[w] = v;
  __syncthreads();
  float s = 0.0f;
  if (w == 0) {
    s = red[lane];
#pragma unroll
    for (int off = 16; off > 0; off >>= 1) s += __shfl_xor(s, off, 32);
    if (lane == 0) red[0] = s;
  }
  __syncthreads();
  s = red[0];
  __syncthreads();
  return s;
}

__global__ __launch_bounds__(1024) void power_iter(const float* __restrict__ G,
                                                   float* __restrict__ scal) {
  __shared__ float v[KATOMS];
  __shared__ float red[32];
  const int t = threadIdx.x;
  v[t] = rsqrtf((float)KATOMS);
  __syncthreads();
  for (int it = 0; it < 20; ++it) {
    float w = 0.0f;
    for (int k = 0; k < KATOMS; ++k) w += G[(size_t)k * KATOMS + t] * v[k]; // symmetric
    float n2 = block_reduce_sum_1024(w * w, red);
    v[t] = w * (1.0f / (sqrtf(n2) + 1e-12f));
    __syncthreads();
  }
  float w = 0.0f;
  for (int k = 0; k < KATOMS; ++k) w += G[(size_t)k * KATOMS + t] * v[k];
  float L = block_reduce_sum_1024(v[t] * w, red);
  if (t == 0) { scal[S_L] = L; scal[S_STEP] = 1.0f / L; scal[S_THR] = ALPHA / L; }
}

// ---------------- fused FISTA step via TDM + WMMA ----------------
// G2 and y live in pair-interleaved global layout, so each K-slab tile is 16
// linear rows: fetched with one tensor_load_to_lds per matrix, double-buffered
// (issue slab i+1, s_wait_tensorcnt for slab i, barrier, WMMA) so the DMA
// overlaps the matrix math. LDS row pad (2 dwords) keeps fragments b64-aligned.
__global__ __launch_bounds__(256) void fista_step(const float* __restrict__ G2,
                                                  const float* __restrict__ yin,
                                                  const float* __restrict__ zin,
                                                  const float* __restrict__ dtx,
                                                  float* __restrict__ yout,
                                                  float* __restrict__ zout,
                                                  const float* __restrict__ scal,
                                                  float mom) {
  __shared__ float As2[2][NPAIR * SA];
  __shared__ float Bs2[2][NPAIR * SB];
  const int tid   = threadIdx.x;
  const int tileM = (blockIdx.x & 7) * BM;
  const int tileN = (blockIdx.x >> 3) * BN;
  const int lane  = tid & 31, wave = tid >> 5;
  v8f acc[4] = {};

  const unsigned ldsA0 = (unsigned)(size_t)&As2[0][0];
  const unsigned ldsA1 = (unsigned)(size_t)&As2[1][0];
  const unsigned ldsB0 = (unsigned)(size_t)&Bs2[0][0];
  const unsigned ldsB1 = (unsigned)(size_t)&Bs2[1][0];
  const unsigned long long gA =
      (unsigned long long)(size_t)G2 + (unsigned long long)(8 * tileM);
  const unsigned long long gB =
      (unsigned long long)(size_t)yin + (unsigned long long)(8 * tileN);
  const unsigned long long slabA = (unsigned long long)NPAIR * (2 * KATOMS) * 4ull;
  const unsigned long long slabB = (unsigned long long)NPAIR * (2 * NPATCH) * 4ull;
  // A: 256-elem rows (pad interval code 7 = 256 dw), B: 128-elem rows (code 6);
  // pad_amount code 1 = 2 dwords -> LDS pitches SA/SB.
  const v8i g1A = tdm_group1(2 * KATOMS, KATOMS / 2, 2 * BM, NPAIR, 2 * KATOMS, 7u, 1u);
  const v8i g1B = tdm_group1(2 * NPATCH, KATOMS / 2, 2 * BN, NPAIR, 2 * NPATCH, 6u, 1u);
  const v4i gz  = {0, 0, 0, 0};
  const v8i gz8 = {0, 0, 0, 0, 0, 0, 0, 0};

  if (wave == 0) {                                     // prologue: slab 0
    __builtin_amdgcn_tensor_load_to_lds(tdm_group0(ldsA0, gA), g1A, gz, gz, gz8, 0);
    __builtin_amdgcn_tensor_load_to_lds(tdm_group0(ldsB0, gB), g1B, gz, gz, gz8, 0);
  }
  const int NSLAB = KATOMS / BK;                       // 32
  for (int i = 0; i < NSLAB; ++i) {
    __syncthreads();                                   // prev slab's LDS reads done
    if (wave == 0) {
      if (i + 1 < NSLAB) {
        const int nb = (i + 1) & 1;
        __builtin_amdgcn_tensor_load_to_lds(
            tdm_group0(nb ? ldsA1 : ldsA0, gA + (unsigned long long)(i + 1) * slabA),
            g1A, gz, gz, gz8, 0);
        __builtin_amdgcn_tensor_load_to_lds(
            tdm_group0(nb ? ldsB1 : ldsB0, gB + (unsigned long long)(i + 1) * slabB),
            g1B, gz, gz, gz8, 0);
        __builtin_amdgcn_s_wait_tensorcnt(2);          // slab i complete, i+1 in flight
      } else {
        __builtin_amdgcn_s_wait_tensorcnt(0);
      }
    }
    __syncthreads();
    wmma_tile2(As2[i & 1], Bs2[i & 1], acc, wave * 16, lane);
  }

  const float step = scal[S_STEP];
  const float thr  = scal[S_THR];
  const int mh = (lane >> 4) << 3;
#pragma unroll
  for (int s = 0; s < 4; ++s)
#pragma unroll
    for (int r = 0; r < 8; ++r) {
      const int gm = tileM + wave * 16 + r + mh;
      const int gn = tileN + s * 16 + (lane & 15);
      const size_t idx  = (size_t)gm * NPATCH + gn;                          // row-major (z, DtX)
      const size_t yidx = (size_t)(gm >> 1) * (2 * NPATCH) + 2 * gn + (gm & 1); // interleaved y
      const float c = acc[s][r];
      const float v = yin[yidx] - step * (c - dtx[idx]);
      const float a = fabsf(v) - thr;
      const float z = (a > 0.0f) ? copysignf(a, v) : 0.0f;
      zout[idx]  = z;
      yout[yidx] = z + mom * (z - zin[idx]);
    }
}

// ---------------- per-patch top-10 + 8-bit quantize + scatter + L0 stats ----------------
__device__ __forceinline__ float quant_coef(float z) {
  z = fminf(fmaxf(z, -0.55f), 0.55f);
  float u = (z + 0.55f) * (1.0f / 1.1f);
  u = rintf(u * 256.0f) * (1.0f / 256.0f);   // round-half-even like jnp.round
  return u * 1.1f - 0.55f;
}

__global__ __launch_bounds__(256) void topk_quant(const float* __restrict__ z,
                                                  float* __restrict__ topv,
                                                  int* __restrict__ topi,
                                                  float* __restrict__ ids,
                                                  float* __restrict__ scal,
                                                  int* __restrict__ hist) {
  const int lane = threadIdx.x & 31;
  const int wave = threadIdx.x >> 5;
  const int n = blockIdx.x * 8 + wave;                 // one wave per patch
  float vals[32];
#pragma unroll
  for (int i = 0; i < 32; ++i)
    vals[i] = z[(size_t)(i * 32 + lane) * NPATCH + n];
  const int b = n >> 10, p = n & 1023;
  int cnt = 0;
  float sabs = 0.0f;
  for (int r = 0; r < TOPK; ++r) {
    float best = -1.0f, bv = 0.0f;
    int bi = 0;
#pragma unroll
    for (int i = 0; i < 32; ++i) {
      float a = fabsf(vals[i]);
      if (a > best) { best = a; bi = i; bv = vals[i]; }
    }
    float rb = best, rv = bv;
    int rl = lane, ri = bi;
#pragma unroll
    for (int off = 16; off > 0; off >>= 1) {           // wave argmax
      float ob = __shfl_xor(rb, off, 32);
      float ov = __shfl_xor(rv, off, 32);
      int   ol = __shfl_xor(rl, off, 32);
      int   oi = __shfl_xor(ri, off, 32);
      bool take = (ob > rb) || (ob == rb && ol < rl);
      if (take) { rb = ob; rv = ov; rl = ol; ri = oi; }
    }
    if (lane == rl) {                                  // winner removes its element
#pragma unroll
      for (int i = 0; i < 32; ++i)
        if (i == ri) vals[i] = 0.0f;
    }
    const int kidx = ri * 32 + rl;
    const float q = quant_coef(rv);
    if (lane == 0) {
      topv[n * TOPK + r] = q;
      topi[n * TOPK + r] = kidx;
      ids[((size_t)b * KATOMS + kidx) * 1024 + p] = q;
      if (q != 0.0f) { cnt++; sabs += fabsf(q); }
    }
  }
  if (lane == 0) {
    atomicAdd(&scal[S_ABSZ], sabs);
    atomicAdd(&scal[S_SUML0], (float)cnt);
    atomicAdd(&hist[cnt], 1);
  }
}

// ---------------- sparse reconstruction + fused qd reduction ----------------
__global__ __launch_bounds__(128) void reconstruct(const float* __restrict__ DnT,
                                                   const float* __restrict__ topv,
                                                   const int* __restrict__ topi,
                                                   const float* __restrict__ Xf,
                                                   float* __restrict__ out0,
                                                   float* __restrict__ scal) {
  __shared__ float sv[TOPK];
  __shared__ int   si[TOPK];
  __shared__ float red[4];
  const int n = blockIdx.x, d = threadIdx.x;
  if (d < TOPK) { sv[d] = topv[n * TOPK + d]; si[d] = topi[n * TOPK + d]; }
  __syncthreads();
  float acc = 0.0f;
#pragma unroll
  for (int j = 0; j < TOPK; ++j) acc += sv[j] * DnT[(size_t)si[j] * DIM + d];
  const int b = n >> 10, p = n & 1023;
  out0[(size_t)b * (DIM * 1024) + (size_t)d * 1024 + p] = acc;   // (B,C,H,W)
  float diff = acc - Xf[(size_t)n * DIM + d];
  float e = diff * diff;
#pragma unroll
  for (int off = 16; off > 0; off >>= 1) e += __shfl_xor(e, off, 32);
  if ((d & 31) == 0) red[d >> 5] = e;
  __syncthreads();
  if (d == 0) atomicAdd(&scal[S_QD], red[0] + red[1] + red[2] + red[3]);
}

// ---------------- scalar outputs ----------------
__global__ void finalize(const float* __restrict__ scal, const int* __restrict__ hist,
                         float* __restrict__ out) {
  if (threadIdx.x != 0 || blockIdx.x != 0) return;
  const float qd = scal[S_QD] / (float)((size_t)NPATCH * DIM);
  out[OFF_QDE]   = qd;
  out[OFF_QDD]   = qd;
  out[OFF_STEPS] = (float)NSTEPS;
  out[OFF_MEAND] = scal[S_ABSD] / (float)(DIM * KATOMS);
  out[OFF_MEANZ] = scal[S_ABSZ] / (float)KN;
  out[OFF_NORMZ] = scal[S_SUML0] / (float)NPATCH;
  int cum = 0, ans = 0;
  for (int c = 63; c >= 0; --c) {                       // 163rd-largest L0
    cum += hist[c];
    if (cum >= 163) { ans = c; break; }
  }
  out[OFF_TOPP]  = (float)ans;
  out[OFF_NZERO] = (float)hist[0];
}

// ---------------- host orchestration ----------------
extern "C" void kernel_launch(void* const* d_in, const int* in_sizes, int n_in,
                              void* d_out, int out_size, void* d_ws, size_t ws_size,
                              hipStream_t stream) {
  const float* x = (const float*)d_in[0];
  const float* D = (const float*)d_in[1];
  float* out = (float*)d_out;
  float* wsf = (float*)d_ws;

  // workspace carve-out (floats); ~355 MB, dominated by 5x 64MB K*N buffers
  float* scal = wsf;                                    // 64 scalars
  int*   hist = (int*)(wsf + 64);                       // 64 bins
  float* Xf   = wsf + 128;                              // N*DIM
  float* Dn   = Xf  + (size_t)NPATCH * DIM;             // DIM*K (d-major)
  float* DnT  = Dn  + (size_t)DIM * KATOMS;             // K*DIM (k-major)
  float* Gm   = DnT + (size_t)DIM * KATOMS;             // K*K row-major (power iter)
  float* G2   = Gm  + (size_t)KATOMS * KATOMS;          // K*K pair-interleaved (TDM)
  float* DtX  = G2  + (size_t)KATOMS * KATOMS;          // K*N row-major
  float* y0   = DtX + KN;                               // pair-interleaved
  float* y1   = y0  + KN;                               // pair-interleaved
  float* z0   = y1  + KN;                               // row-major
  float* z1   = z0  + KN;                               // row-major
  float* topv = z1  + KN;                               // N*TOPK
  int*   topi = (int*)(topv + (size_t)NPATCH * TOPK);   // N*TOPK

  // zero state used this call (harness does not re-zero between replays)
  zero_f32<<<4, 256, 0, stream>>>(scal, 128);
  zero_f32<<<2048, 256, 0, stream>>>(y0, KN);
  zero_f32<<<2048, 256, 0, stream>>>(z0, KN);
  zero_f32<<<2048, 256, 0, stream>>>(out + OFF_IDS, KN);

  permute_in<<<8192, 256, 0, stream>>>(x, Xf);
  normalize_dict<<<4, 256, 0, stream>>>(D, Dn, DnT, scal);
  gemm_gram<<<8 * (KATOMS / BN), 256, 0, stream>>>(Dn, Gm, G2);
  gemm_dtx<<<8 * (NPATCH / BN), 256, 0, stream>>>(Dn, Xf, DtX);
  power_iter<<<1, 1024, 0, stream>>>(Gm, scal);

  // FISTA: momentum coefficients are a pure host-side scalar recurrence
  float t = 1.0f;
  float *ya = y0, *yb = y1, *za = z0, *zb = z1;
  for (int j = 0; j < NSTEPS; ++j) {
    float tn  = 0.5f * (1.0f + sqrtf(1.0f + 4.0f * t * t));
    float mom = (t - 1.0f) / tn;
    fista_step<<<8 * (NPATCH / BN), 256, 0, stream>>>(G2, ya, za, DtX, yb, zb, scal, mom);
    t = tn;
    float* tmp = ya; ya = yb; yb = tmp;
    tmp = za; za = zb; zb = tmp;
  }
  // after the final swap, za holds the last-written z

  topk_quant<<<NPATCH / 8, 256, 0, stream>>>(za, topv, topi, out + OFF_IDS, scal, hist);
  reconstruct<<<NPATCH, 128, 0, stream>>>(DnT, topv, topi, Xf, out, scal);
  finalize<<<1, 64, 0, stream>>>(scal, hist, out);
}